// SkipARNN_46952582480193
// MI455X (gfx1250) — compile-verified
//
#include <hip/hip_runtime.h>
#include <math.h>

#define HDIM 4096
#define KSEG 16
#define SEGK (HDIM / KSEG)                 // 256
#define ROWS_PER_BLOCK 128                 // 8 waves * 16 rows
#define ROWBLOCKS (HDIM / ROWS_PER_BLOCK)  // 32
#define NMAX 16
#define EPSF 0.01f

typedef __attribute__((ext_vector_type(2))) float v2f;
typedef __attribute__((ext_vector_type(8))) float v8f;
typedef int v4i_b128 __attribute__((vector_size(16)));   // matches builtin param pointee

#if defined(__AMDGCN__) && __has_builtin(__builtin_amdgcn_global_load_async_to_lds_b128) && \
    __has_builtin(__builtin_amdgcn_s_wait_asynccnt)
#define USE_ASYNC_LDS 1
#else
#define USE_ASYNC_LDS 0
#endif

// ---------------------------------------------------------------------------
// Partial matvec: partial[kseg][m] = sum_{k in seg} W[m][k] * vec[k]
// Implemented with V_WMMA_F32_16X16X4_F32 (exact fp32).
//   A tile: 16 rows of W, K=4 chunk.  Lane layout (ISA 7.12.2):
//     lanes 0-15  : VGPR0=K0, VGPR1=K1 for M=lane
//     lanes 16-31 : VGPR0=K2, VGPR1=K3 for M=lane-16
//   B tile: vector chunk broadcast across all 16 N columns (mirror of A's K split).
//   C: column N=0 lives in lane 0 (M=0..7) and lane 16 (M=8..15).
// Vector segment staged into LDS via GLOBAL_LOAD_ASYNC_TO_LDS_B128 (ASYNCcnt).
// Gated uniformly so EXEC is all-1s at every WMMA.
// ---------------------------------------------------------------------------
__global__ __launch_bounds__(256)
void wmma_matvec_kernel(const float* __restrict__ W,
                        const float* __restrict__ vec0,
                        const float* __restrict__ vec1,
                        const int*   __restrict__ selPtr,
                        float*       __restrict__ partial,
                        const int*   __restrict__ gatePtr,
                        int gateSkipVal)
{
    if (gatePtr && *gatePtr == gateSkipVal) return;   // uniform early-out
    const float* vec = vec0;
    if (selPtr && *selPtr != 0) vec = vec1;

    __shared__ __align__(16) float sv[SEGK];
    const int kseg   = blockIdx.x & (KSEG - 1);
    const int rowblk = blockIdx.x / KSEG;
    const int kbase  = kseg * SEGK;

#if USE_ASYNC_LDS
    // 64 lanes x 16B async copies: global -> LDS without a VGPR round-trip.
    if (threadIdx.x < SEGK / 4) {
        const float* gsrc = vec + kbase + threadIdx.x * 4;
        __attribute__((address_space(1))) void* g1 =
            (__attribute__((address_space(1))) void*)(void*)const_cast<float*>(gsrc);
        __attribute__((address_space(3))) void* l3 =
            (__attribute__((address_space(3))) void*)(void*)&sv[threadIdx.x * 4];
        __builtin_amdgcn_global_load_async_to_lds_b128(
            (__attribute__((address_space(1))) v4i_b128*)g1,
            (__attribute__((address_space(3))) v4i_b128*)l3,
            /*imm offset*/0, /*cpol*/0);
    }
    __builtin_amdgcn_s_wait_asynccnt(0);
    __syncthreads();
#else
    for (int i = threadIdx.x; i < SEGK; i += 256) sv[i] = vec[kbase + i];
    __syncthreads();
#endif

    const int wave  = threadIdx.x >> 5;
    const int lane  = threadIdx.x & 31;
    const int m     = lane & 15;
    const int khalf = (lane >> 4) << 1;               // 0 or 2
    const int row0  = rowblk * ROWS_PER_BLOCK + wave * 16;
    const float* wrow = W + (size_t)(row0 + m) * HDIM + kbase;

    v8f acc = {};
    for (int k = 0; k < SEGK; k += 4) {
        v2f a = *reinterpret_cast<const v2f*>(wrow + k + khalf);     // W[m][k+khalf..+1]
        v2f b = *reinterpret_cast<const v2f*>(&sv[k + khalf]);       // vec chunk (bcast over N)
        acc = __builtin_amdgcn_wmma_f32_16x16x4_f32(false, a, false, b,
                                                    (short)0, acc, false, false);
    }
    // Extract column N=0
    if ((lane & 15) == 0) {
        const int mbase = row0 + ((lane >> 4) << 3);
        float* p = partial + (size_t)kseg * HDIM + mbase;
        #pragma unroll
        for (int r = 0; r < 8; ++r) p[r] = acc[r];
    }
}

__global__ __launch_bounds__(256)
void init_kernel(const float* __restrict__ x, const float* __restrict__ h0,
                 float* __restrict__ xin, float* __restrict__ stf, int* __restrict__ sti)
{
    int i = blockIdx.x * 256 + threadIdx.x;
    if (i < HDIM) xin[i] = (i == 0) ? 0.0f : x[i - 1];   // xin = [skip_flag=0, x]
    if (i == 0) {
        float a = h0[0];
        stf[0] = a;                               // acc carry
        stf[1] = 0.0f;                            // ponder counter
        stf[2] = 0.0f;                            // last_acc at halting step
        sti[3] = (a >= 1.0f - EPSF) ? 1 : 0;      // halted (== !valid)
        sti[4] = 0;                               // selected s-buffer index
    }
}

__global__ __launch_bounds__(256)
void combine_c_kernel(const float* __restrict__ partial, const float* __restrict__ b_ih,
                      const float* __restrict__ b_hh, float* __restrict__ c)
{
    int i = blockIdx.x * 256 + threadIdx.x;
    float a = b_ih[i] + b_hh[i];
    #pragma unroll
    for (int s = 0; s < KSEG; ++s) a += partial[s * HDIM + i];
    c[i] = a;                                     // c = W_ih@[0,x] + b_ih + b_hh
}

__global__ __launch_bounds__(256)
void activation_kernel(const float* __restrict__ c, const float* __restrict__ partial,
                       const float* __restrict__ W_ih, float* __restrict__ s_out,
                       int first, const int* __restrict__ halted)
{
    if (*halted) return;
    int i = blockIdx.x * 256 + threadIdx.x;
    float a = c[i];
    if (first) a += W_ih[(size_t)i * HDIM];       // + W_ih[:,0] at step 0 (skip flag = 1)
    #pragma unroll
    for (int s = 0; s < KSEG; ++s) a += partial[s * HDIM + i];
    s_out[i] = tanhf(a);
}

__global__ __launch_bounds__(256)
void finalize_kernel(const float* __restrict__ s_new, const float* __restrict__ W_halt,
                     const float* __restrict__ b_halt, float* __restrict__ stf,
                     int* __restrict__ sti, int t)
{
    if (sti[3]) return;                           // already halted (uniform)
    __shared__ float red[256];
    float sum = 0.0f;
    for (int i = threadIdx.x; i < HDIM; i += 256) sum += W_halt[i] * s_new[i];
    red[threadIdx.x] = sum;
    __syncthreads();
    for (int off = 128; off > 0; off >>= 1) {
        if ((int)threadIdx.x < off) red[threadIdx.x] += red[threadIdx.x + off];
        __syncthreads();
    }
    if (threadIdx.x == 0) {
        float hd = red[0] + b_halt[0];
        float p  = 2.0f / (1.0f + expf(-hd));     // MU * sigmoid
        float acc_new = stf[0] + p;
        if (acc_new >= 1.0f - EPSF) {             // this is the halting step
            sti[3] = 1;
            sti[4] = (t + 1) & 1;                 // buffer holding s at halting step
            stf[2] = acc_new;
        } else {
            stf[1] += 1.0f;                       // ponder += 1 (valid, not yet halted)
        }
        stf[0] = acc_new;
    }
}

__global__ __launch_bounds__(256)
void combine_out_kernel(const float* __restrict__ partial, const float* __restrict__ b_out,
                        const float* __restrict__ y0, const float* __restrict__ s0,
                        const float* __restrict__ h0,
                        const float* __restrict__ sb0, const float* __restrict__ sb1,
                        const float* __restrict__ stf, const int* __restrict__ sti,
                        float* __restrict__ out)
{
    int i = blockIdx.x * 256 + threadIdx.x;       // 0..4095
    const int h0bin  = (h0[0] >= 1.0f - EPSF) ? 1 : 0;
    const int halted = sti[3];

    float y, s;
    if (h0bin) {
        y = y0[i]; s = s0[i];
    } else if (halted) {
        float a = b_out[i];
        #pragma unroll
        for (int sgi = 0; sgi < KSEG; ++sgi) a += partial[sgi * HDIM + i];
        y = a;
        const float* ss = sti[4] ? sb1 : sb0;
        s = ss[i];
    } else {                                      // never halted within N_MAX
        y = 0.0f; s = 0.0f;
    }
    out[i]        = y;
    out[HDIM + i] = s;
    if (i == 0) {
        float ponder = h0bin ? 0.0f : stf[1];
        float last   = h0bin ? h0[0] : (halted ? stf[2] : 0.0f);
        out[2 * HDIM]     = ponder;
        out[2 * HDIM + 1] = last - 1.0f;
    }
}

extern "C" void kernel_launch(void* const* d_in, const int* in_sizes, int n_in,
                              void* d_out, int out_size, void* d_ws, size_t ws_size,
                              hipStream_t stream) {
    const float* x      = (const float*)d_in[0];
    const float* s0     = (const float*)d_in[1];
    const float* y0     = (const float*)d_in[2];
    const float* h0     = (const float*)d_in[3];
    const float* W_ih   = (const float*)d_in[4];
    const float* b_ih   = (const float*)d_in[5];
    const float* W_hh   = (const float*)d_in[6];
    const float* b_hh   = (const float*)d_in[7];
    const float* W_halt = (const float*)d_in[8];
    const float* b_halt = (const float*)d_in[9];
    const float* W_out  = (const float*)d_in[10];
    const float* b_out  = (const float*)d_in[11];
    float* out = (float*)d_out;

    float* ws   = (float*)d_ws;
    float* xin  = ws;                              // [4096]
    float* c    = ws + HDIM;                       // [4096]
    float* sb0  = ws + 2 * HDIM;                   // [4096] ping
    float* sb1  = ws + 3 * HDIM;                   // [4096] pong
    float* part = ws + 4 * HDIM;                   // [KSEG][4096]
    float* stf  = ws + 4 * HDIM + KSEG * HDIM;     // state scalars
    int*   sti  = (int*)stf;

    const dim3 mvGrid(ROWBLOCKS * KSEG);           // 512 blocks
    const dim3 mvBlk(256);

    init_kernel<<<16, 256, 0, stream>>>(x, h0, xin, stf, sti);

    // c = W_ih @ [0,x]  (always runs; combined with both biases)
    wmma_matvec_kernel<<<mvGrid, mvBlk, 0, stream>>>(W_ih, xin, nullptr, nullptr,
                                                     part, nullptr, 0);
    combine_c_kernel<<<16, 256, 0, stream>>>(part, b_ih, b_hh, c);

    // ACT loop, unrolled to N_MAX fixed launches, gated on device-side `halted`
    for (int t = 0; t < NMAX; ++t) {
        const float* vin  = (t == 0) ? s0 : ((t & 1) ? sb1 : sb0);
        float*       vout = ((t + 1) & 1) ? sb1 : sb0;
        wmma_matvec_kernel<<<mvGrid, mvBlk, 0, stream>>>(W_hh, vin, nullptr, nullptr,
                                                         part, &sti[3], /*skip if halted*/1);
        activation_kernel<<<16, 256, 0, stream>>>(c, part, W_ih, vout,
                                                  (t == 0) ? 1 : 0, &sti[3]);
        finalize_kernel<<<1, 256, 0, stream>>>(vout, W_halt, b_halt, stf, sti, t);
    }

    // y = W_out @ s_halt  (runs only if a halting step occurred; selector is device-side)
    wmma_matvec_kernel<<<mvGrid, mvBlk, 0, stream>>>(W_out, sb0, sb1, &sti[4],
                                                     part, &sti[3], /*skip if NOT halted*/0);
    combine_out_kernel<<<16, 256, 0, stream>>>(part, b_out, y0, s0, h0,
                                               sb0, sb1, stf, sti, out);
}